// RoIPointQueryStack_26259430048601
// MI455X (gfx1250) — compile-verified
//
#include <hip/hip_runtime.h>
#include <stdint.h>

#define TPB   1024
#define WAVES (TPB / 32)
#define PPT   2                 // points per thread per chunk
#define CHUNK (TPB * PPT)       // 2048 points per chunk

#if __has_builtin(__builtin_amdgcn_global_load_async_to_lds_b32)
#define HAVE_ASYNC 1
#else
#define HAVE_ASYNC 0
#endif

#if __has_builtin(__builtin_amdgcn_s_wait_asynccnt)
#define WAIT_ASYNC(nimm) __builtin_amdgcn_s_wait_asynccnt(nimm)
#else
#define WAIT_ASYNC(nimm) asm volatile("s_wait_asynccnt " #nimm ::: "memory")
#endif

__device__ __forceinline__ unsigned ballot32(bool p) {
#if __has_builtin(__builtin_amdgcn_ballot_w32)
  return __builtin_amdgcn_ballot_w32(p);
#else
  return (unsigned)__ballot(p);
#endif
}

__global__ __launch_bounds__(TPB) void RoIPointQueryStack_kernel(
    const float* __restrict__ points,   // (N,3)
    const int*   __restrict__ pcnt,     // (B,)
    const float* __restrict__ boxes,    // (B*M,7)
    int*         __restrict__ out,      // (B*M,K)
    int M, int K) {
  const int t    = threadIdx.x;
  const int lane = t & 31;
  const int w    = t >> 5;
  const int bid   = blockIdx.y;            // batch   (no integer division)
  const int boxid = bid * M + blockIdx.x;  // flat box id

  // Batch start offset (batch ids are contiguous/sorted per setup_inputs).
  int start = 0;
  for (int b = 0; b < bid; ++b) start += pcnt[b];
  const int n = pcnt[bid];

  // Box parameters, enlarged by 2*EXTRA = 2.0 on each size dim.
  const float* bx = boxes + (size_t)boxid * 7;
  const float cx = bx[0], cy = bx[1], cz = bx[2];
  const float hdx = (bx[3] + 2.0f) * 0.5f;
  const float hdy = (bx[4] + 2.0f) * 0.5f;
  const float hdz = (bx[5] + 2.0f) * 0.5f;
  const float hh  = bx[6];
  const float ch = cosf(hh);
  const float sh = sinf(hh);

  __shared__ int scnt[WAVES];
#if HAVE_ASYNC
  __shared__ unsigned sbuf[2][CHUNK * 3];  // double-buffered chunks (2 x 24 KB)

  auto stage = [&](int c, int buf) {
    int idx0 = c * CHUNK + t * PPT;                  // even point index
    int maxb = (n >= PPT) ? n - PPT : 0;
    int cidx = idx0 < maxb ? idx0 : maxb;            // clamp (uniform issue)
    int* src = (int*)((const unsigned*)points + (size_t)(start + cidx) * 3);
    int* dst = (int*)&sbuf[buf][t * (3 * PPT)];
    __builtin_amdgcn_global_load_async_to_lds_b32(src + 0, dst + 0, 0, 0);
    __builtin_amdgcn_global_load_async_to_lds_b32(src + 1, dst + 1, 0, 0);
    __builtin_amdgcn_global_load_async_to_lds_b32(src + 2, dst + 2, 0, 0);
    __builtin_amdgcn_global_load_async_to_lds_b32(src + 3, dst + 3, 0, 0);
    __builtin_amdgcn_global_load_async_to_lds_b32(src + 4, dst + 4, 0, 0);
    __builtin_amdgcn_global_load_async_to_lds_b32(src + 5, dst + 5, 0, 0);
  };
#endif

  const int nch = (n + CHUNK - 1) / CHUNK;
  int total = 0;

#if HAVE_ASYNC
  if (nch > 0) stage(0, 0);
#endif

  for (int c = 0; c < nch; ++c) {
    const int idx0 = c * CHUNK + t * PPT;

    float x0, y0, z0, x1, y1, z1;
#if HAVE_ASYNC
    if (c + 1 < nch) {
      stage(c + 1, (c + 1) & 1);       // overlap next chunk's DMA
      WAIT_ASYNC(6);                   // chunk c done, chunk c+1 in flight
    } else {
      WAIT_ASYNC(0);
    }
    {
      const volatile unsigned* pb = &sbuf[c & 1][t * (3 * PPT)];
      x0 = __uint_as_float(pb[0]);
      y0 = __uint_as_float(pb[1]);
      z0 = __uint_as_float(pb[2]);
      x1 = __uint_as_float(pb[3]);
      y1 = __uint_as_float(pb[4]);
      z1 = __uint_as_float(pb[5]);
    }
#else
    {
      int maxb = (n >= PPT) ? n - PPT : 0;
      int cidx = idx0 < maxb ? idx0 : maxb;
      const float* p = points + (size_t)(start + cidx) * 3;
      x0 = p[0]; y0 = p[1]; z0 = p[2];
      x1 = p[3]; y1 = p[4]; z1 = p[5];
    }
#endif

    // Rotated-box containment tests for the consecutive pair.
    const float rx0 = x0 - cx, ry0 = y0 - cy;
    const float lx0 =  rx0 * ch + ry0 * sh;
    const float ly0 = -rx0 * sh + ry0 * ch;
    const float lz0 =  z0 - cz;
    const bool in0 = (idx0 < n) &&
                     (fabsf(lx0) < hdx) && (fabsf(ly0) < hdy) && (fabsf(lz0) < hdz);

    const float rx1 = x1 - cx, ry1 = y1 - cy;
    const float lx1 =  rx1 * ch + ry1 * sh;
    const float ly1 = -rx1 * sh + ry1 * ch;
    const float lz1 =  z1 - cz;
    const bool in1 = (idx0 + 1 < n) &&
                     (fabsf(lx1) < hdx) && (fabsf(ly1) < hdy) && (fabsf(lz1) < hdz);

    // wave32 ordered compaction over interleaved pairs:
    // point order within chunk is (lane0:p0,p1)(lane1:p2,p3)... so the
    // even point's rank counts BOTH masks strictly below this lane.
    const unsigned mask0 = ballot32(in0);
    const unsigned mask1 = ballot32(in1);
    const unsigned below = (1u << lane) - 1u;
    const int r0 = __popc(mask0 & below) + __popc(mask1 & below);
    const int r1 = r0 + (in0 ? 1 : 0);

    __syncthreads();                               // protect scnt from prior reads
    if (lane == 0) scnt[w] = __popc(mask0) + __popc(mask1);
    __syncthreads();

    // Cross-wave exclusive scan of the 32 wave counts (one value per lane).
    int v = scnt[lane];
    #pragma unroll
    for (int d = 1; d < 32; d <<= 1) {
      int u = __shfl_up(v, d, 32);
      if (lane >= d) v += u;
    }
    const int sum   = __shfl(v, 31, 32);
    const int basew = (w == 0) ? 0 : __shfl(v, w - 1, 32);
    const int base  = total + basew;

    if (in0) {
      const int s0 = base + r0;
      if (s0 < K) out[(size_t)boxid * K + s0] = start + idx0;
    }
    if (in1) {
      const int s1 = base + r1;
      if (s1 < K) out[(size_t)boxid * K + s1] = start + idx0 + 1;
    }
    total += sum;                                  // uniform across block
    if (total >= K) break;                         // uniform early exit
  }

  // Fill unfilled slots with -1 (filled count == min(total, K)).
  for (int j = total + t; j < K; j += TPB) out[(size_t)boxid * K + j] = -1;
}

extern "C" void kernel_launch(void* const* d_in, const int* in_sizes, int n_in,
                              void* d_out, int out_size, void* d_ws, size_t ws_size,
                              hipStream_t stream) {
  (void)n_in; (void)d_ws; (void)ws_size;
  const float* points = (const float*)d_in[0];
  // d_in[1] (points_batch_id) unused: batches are contiguous, offsets from pcnt.
  const int*   pcnt   = (const int*)d_in[2];
  const float* boxes  = (const float*)d_in[3];
  int*         out    = (int*)d_out;

  const int B = in_sizes[2];
  const int M = (B > 0) ? in_sizes[3] / (7 * B) : 0;
  const int nbox = B * M;
  const int K = (nbox > 0) ? out_size / nbox : 0;
  if (nbox <= 0 || K <= 0) return;

  dim3 grid(M, B, 1);
  RoIPointQueryStack_kernel<<<grid, TPB, 0, stream>>>(points, pcnt, boxes, out, M, K);
}